// PatchLoss_32323923869699
// MI455X (gfx1250) — compile-verified
//
#include <hip/hip_runtime.h>

// PatchLoss on MI455X (gfx1250): conv-as-implicit-GEMM + spectral GEMM on
// V_WMMA_F32_16X16X4_F32 (fp32 matrix path — argmax correctness requires fp32),
// with async global->LDS staging (ASYNCcnt) and TDM tensor_load_to_lds (TENSORcnt).

#define WINDOW     512
#define STEP       256
#define NSAMP      32768
#define NFRAMES    127
#define FPAD       128      // frames padded to 128 rows (zeros) -> full-EXEC WMMA
#define BATCH      4
#define NSPEC      32769    // rfft bins of 65536-pt padded transform
#define LDS_STRIDE 516      // 512 + 4 pad: 16-row column reads hit 16 distinct banks
#define TWIN       640      // staged target window per block: [tbase-512, tbase+128)

typedef float v2f __attribute__((ext_vector_type(2)));
typedef float v8f __attribute__((ext_vector_type(8)));
typedef unsigned uint32x4 __attribute__((ext_vector_type(4)));
typedef int int32x8 __attribute__((ext_vector_type(8)));
typedef int int32x4 __attribute__((ext_vector_type(4)));
typedef __attribute__((address_space(3))) float lds_f32;

// workspace layout (bytes)
#define WS_FRAMES 0                                    // BATCH*FPAD*WINDOW f32   (1 MB)
#define WS_KEYS   (BATCH*FPAD*WINDOW*4)                // BATCH*FPAD u64          (4 KB)
#define WS_TPOS   (WS_KEYS + BATCH*FPAD*8)             // BATCH*FPAD i32          (2 KB)
#define WS_SPEC   (WS_TPOS + BATCH*FPAD*4)             // BATCH*NSPEC*2 f32       (1 MB)

__global__ void k_init(unsigned long long* keys, float* out) {
    int i = blockIdx.x * blockDim.x + threadIdx.x;
    if (i < BATCH * FPAD) keys[i] = 0ull;   // 0 == most-negative ordered float key
    if (i == 0) out[0] = 0.0f;
}

// Build Hamming-windowed frames, zero-pad row 127 (FPAD) for full-EXEC WMMA.
__global__ void k_frames(const float* __restrict__ recon, float* __restrict__ frames) {
    int bf = blockIdx.x;                 // b*FPAD + f
    int b = bf >> 7, f = bf & 127;
    int tau = threadIdx.x;
    #pragma unroll
    for (int e = 0; e < 2; ++e) {
        float v = 0.0f;
        if (f < NFRAMES) {
            // periodic hamming: 0.54 - 0.46*cos(2*pi*k/512)
            float ham = 0.54f - 0.46f * __cosf(6.28318530718e0f * (float)tau * (1.0f / 512.0f));
            v = recon[b * NSAMP + f * STEP + tau] * ham;
        }
        frames[bf * WINDOW + tau] = v;
        tau += 256;
    }
    // cluster-wide sync point (S_NOP when dispatched without clusters)
    __builtin_amdgcn_s_cluster_barrier();
}

// Implicit-GEMM cross-correlation fm[f,t] = sum_tau w[f,tau]*tgt[t-tau], fused argmax.
// One wave per 16x16 (frame x t) tile; K = 512 taps in steps of 4 (wmma f32 16x16x4).
// A-tile staged via async global->LDS DMA; B window staged once -> branch-free K loop.
__global__ __launch_bounds__(256) void k_conv_argmax(
        const float* __restrict__ frames, const float* __restrict__ target,
        unsigned long long* __restrict__ keys) {
    __shared__ float ldsA[16 * LDS_STRIDE];
    __shared__ float ldsT[TWIN];
    const int b   = blockIdx.z;
    const int fb  = blockIdx.y;          // frame block (16 frames)
    const int tid = threadIdx.x;

    // gfx1250 cluster-id read (always 0 for non-clustered dispatch; never taken)
    if (__builtin_amdgcn_cluster_id_x() > 16) return;

    const float* tgt   = target + b * NSAMP;
    const int    tbase = blockIdx.x * 128;   // first t of this block (8 tiles x 16)

    // ---- async-stage 16x512 A tile into LDS (b128 chunks, ASYNCcnt tracked) ----
    {
        const float* src = frames + ((b * FPAD) + fb * 16) * WINDOW;
        for (int i = tid; i < (16 * WINDOW) / 4; i += 256) {   // 2048 16B chunks
            int r = i >> 7, c = (i & 127) << 2;
            lds_f32* lp = (lds_f32*)&ldsA[r * LDS_STRIDE + c];
            unsigned long long ga = (unsigned long long)(const void*)(src + r * WINDOW + c);
            asm volatile("global_load_async_to_lds_b128 %0, %1, off"
                         :: "v"(lp), "v"(ga) : "memory");
        }
    }
    // ---- stage target window [tbase-512, tbase+128) with zero left-padding ----
    for (int i = tid; i < TWIN; i += 256) {
        int g = tbase - 512 + i;
        ldsT[i] = (g >= 0) ? tgt[g] : 0.0f;
    }
    asm volatile("s_wait_asynccnt 0x0" ::: "memory");
    __syncthreads();

    const int wave = tid >> 5, lane = tid & 31;
    const int lh = lane >> 4, lm = lane & 15;     // half, lane-in-half
    const int tloc = 512 + wave * 16 + lm;        // this lane's t in ldsT coords
    const int t    = tbase + wave * 16 + lm;      // absolute t

    v8f c8 = {};
    for (int kt = 0; kt < WINDOW; kt += 4) {
        // A 16x4: lane=M, vgpr v -> K = v + 2*lh
        int tau0 = kt + 2 * lh;
        v2f a, bb;
        a.x = ldsA[lm * LDS_STRIDE + tau0];
        a.y = ldsA[lm * LDS_STRIDE + tau0 + 1];
        // B 4x16: B[K][N] = tgt[t - (kt+K)] from the staged window (branch-free)
        bb.x = ldsT[tloc - tau0];
        bb.y = ldsT[tloc - tau0 - 1];
        c8 = __builtin_amdgcn_wmma_f32_16x16x4_f32(false, a, false, bb,
                                                   (short)0, c8, false, false);
    }

    // fused argmax: key = ordered(value)<<32 | ~t  (max-key => max value, then min t,
    // matching jnp.argmax first-occurrence tie-break; positive ortho scale dropped).
    #pragma unroll
    for (int r = 0; r < 8; ++r) {
        unsigned u = __float_as_uint(c8[r]);
        u = (u & 0x80000000u) ? ~u : (u | 0x80000000u);
        unsigned long long key =
            ((unsigned long long)u << 32) | (unsigned)(0xFFFFFFFFu - (unsigned)t);
        #pragma unroll
        for (int off = 1; off < 16; off <<= 1) {   // reduce within 16-lane half
            unsigned lo = (unsigned)key, hi = (unsigned)(key >> 32);
            lo = __shfl_xor(lo, off, 32);
            hi = __shfl_xor(hi, off, 32);
            unsigned long long o = ((unsigned long long)hi << 32) | lo;
            if (o > key) key = o;
        }
        if (lm == 0) {
            int f = fb * 16 + r + 8 * lh;          // C/D layout: M = r + 8*half
            atomicMax(&keys[b * FPAD + f], key);
        }
    }
}

__global__ void k_positions(const unsigned long long* __restrict__ keys,
                            int* __restrict__ tpos) {
    int i = blockIdx.x * blockDim.x + threadIdx.x;
    if (i >= BATCH * FPAD) return;
    tpos[i] = (int)(0xFFFFFFFFu - (unsigned)(keys[i] & 0xFFFFFFFFull));
}

// Spectral GEMM: spec[f,k] = sum_tau w[f,tau] * e^{-2pi i k tau/65536}  (K=512),
// rotate by per-frame fft_shift phase e^{-i pi k t_f/32769}, reduce over frames.
// A-tile staged by the Tensor Data Mover; D# pad fields (2 DW pad / 256 DW) make the
// DMA itself produce the 516-float row stride: elem (r,c) at r*516 + c + 2*(c>>8).
__global__ __launch_bounds__(256) void k_spectra(
        const float* __restrict__ frames, const int* __restrict__ tpos,
        float* __restrict__ S) {
    __shared__ float ldsA[16 * LDS_STRIDE];   // 8256 f32 == padded 16x512 tile
    const int b   = blockIdx.y;
    const int tid = threadIdx.x;
    const int wave = tid >> 5, lane = tid & 31;
    const int lh = lane >> 4, lm = lane & 15;
    const int k = (blockIdx.x * 8 + wave) * 16 + lm;   // this lane's frequency bin

    float sre = 0.0f, sim = 0.0f;

    for (int fbk = 0; fbk < 8; ++fbk) {
        __syncthreads();                                // previous tile fully consumed
        if (tid < 32) {                                 // wave 0 drives the TDM
            unsigned long long ga =
                (unsigned long long)(const void*)(frames + ((b * FPAD) + fbk * 16) * WINDOW);
            unsigned ldsoff = (unsigned)(unsigned long long)(lds_f32*)&ldsA[0];
            uint32x4 g0;
            g0.x = 1u;                                  // count=1 user descriptor
            g0.y = ldsoff;                              // lds_addr
            g0.z = (unsigned)ga;                        // global_addr[31:0]
            g0.w = (unsigned)((ga >> 32) & 0x01FFFFFFull) | (2u << 30); // [56:32], type=2
            int32x8 g1;
            g1[0] = (2 << 16) | (1 << 20) | (7 << 22) | (1 << 25);
            //        ^4B elem   ^pad_en    ^256 DW int  ^2 DW pad
            g1[1] = (int)((unsigned)WINDOW << 16);      // tensor_dim0=512 (lo16)
            g1[2] = (int)(16u << 16);                   // dim0 hi=0; tensor_dim1=16 lo16
            g1[3] = (int)((unsigned)WINDOW << 16);      // dim1 hi=0; tile_dim0=512
            g1[4] = 16;                                 // tile_dim1=16; tile_dim2=0
            g1[5] = WINDOW;                             // tensor_dim0_stride=512
            g1[6] = 0; g1[7] = 0;
            int32x4 gz4 = {0, 0, 0, 0};                 // groups 2/3 unused (2-D tile)
            int32x8 gz8 = {0, 0, 0, 0, 0, 0, 0, 0};     // clang-23 6-arg form extra group
            __builtin_amdgcn_tensor_load_to_lds(g0, g1, gz4, gz4, gz8, 0);
            __builtin_amdgcn_s_wait_tensorcnt(0);
        }
        __syncthreads();

        v8f cre = {}, cim = {};
        for (int kt = 0; kt < WINDOW; kt += 4) {
            int tau0 = kt + 2 * lh;
            int ai = lm * LDS_STRIDE + tau0 + ((tau0 >> 8) << 1);  // TDM-padded layout
            v2f a, bc, bs;
            a.x = ldsA[ai];
            a.y = ldsA[ai + 1];
            // twiddles, exact arg reduction mod 65536
            int m0 = (k * tau0) & 65535;
            int m1 = (k * (tau0 + 1)) & 65535;
            float s0, c0, s1, c1;
            __sincosf((float)m0 * 9.58737992e-5f, &s0, &c0);   // 2*pi/65536
            __sincosf((float)m1 * 9.58737992e-5f, &s1, &c1);
            bc.x = c0;  bc.y = c1;
            bs.x = -s0; bs.y = -s1;
            cre = __builtin_amdgcn_wmma_f32_16x16x4_f32(false, a, false, bc,
                                                        (short)0, cre, false, false);
            cim = __builtin_amdgcn_wmma_f32_16x16x4_f32(false, a, false, bs,
                                                        (short)0, cim, false, false);
        }

        #pragma unroll
        for (int r = 0; r < 8; ++r) {
            int f = fbk * 16 + r + 8 * lh;
            if (f < NFRAMES) {
                // phase = exp(-i*pi*k*t_f/32769); period 65538 in k*t_f
                long long kt64 = (long long)k * (long long)tpos[b * FPAD + f];
                int m2 = (int)(kt64 % 65538);
                float ps, pc;
                __sincosf(-(float)m2 * 9.58708738e-5f, &ps, &pc);  // pi/32769
                sre += cre[r] * pc - cim[r] * ps;
                sim += cre[r] * ps + cim[r] * pc;
            }
        }
    }
    // combine the two frame-halves (same k in lanes l and l+16)
    sre += __shfl_xor(sre, 16, 32);
    sim += __shfl_xor(sim, 16, 32);
    if (lh == 0 && k < NSPEC) {
        S[(b * NSPEC + k) * 2 + 0] = sre * 0.00390625f;  // 1/256: rfft-ortho of 65536
        S[(b * NSPEC + k) * 2 + 1] = sim * 0.00390625f;
    }
}

// Inverse real ortho-DFT of S (Hermitian expansion), crop to n, MSE vs target.
__global__ __launch_bounds__(256) void k_irfft_mse(
        const float* __restrict__ S, const float* __restrict__ target,
        float* __restrict__ out) {
    __shared__ float sRe[256], sIm[256], red[256];
    const int b   = blockIdx.y;
    const int tid = threadIdx.x;
    const int t   = blockIdx.x * 256 + tid;
    const float* Sb = S + b * NSPEC * 2;

    float acc = 0.0f;
    for (int kc = 0; kc < 32768; kc += 256) {
        __syncthreads();
        sRe[tid] = Sb[(kc + tid) * 2 + 0];
        sIm[tid] = Sb[(kc + tid) * 2 + 1];
        __syncthreads();
        for (int j = 0; j < 256; ++j) {
            int kk = kc + j;
            int m = (kk * t) & 65535;
            float sn, cs;
            __sincosf((float)m * 9.58737992e-5f, &sn, &cs);
            float w = (kk == 0) ? 1.0f : 2.0f;        // Hermitian doubling
            acc += w * (sRe[j] * cs - sIm[j] * sn);
        }
    }
    acc += ((t & 1) ? -1.0f : 1.0f) * Sb[32768 * 2 + 0];  // Nyquist bin
    float x = acc * 0.00390625f;                          // 1/256: irfft-ortho

    float e = x - target[b * NSAMP + t];
    red[tid] = e * e;
    __syncthreads();
    for (int s = 128; s > 0; s >>= 1) {
        if (tid < s) red[tid] += red[tid + s];
        __syncthreads();
    }
    if (tid == 0) atomicAdd(out, red[0] * (1.0f / 131072.0f));  // mean over B*n
}

extern "C" void kernel_launch(void* const* d_in, const int* in_sizes, int n_in,
                              void* d_out, int out_size, void* d_ws, size_t ws_size,
                              hipStream_t stream) {
    (void)in_sizes; (void)n_in; (void)out_size; (void)ws_size;
    const float* recon  = (const float*)d_in[0];
    const float* target = (const float*)d_in[1];
    float* out = (float*)d_out;

    char* ws = (char*)d_ws;
    float*              frames = (float*)(ws + WS_FRAMES);
    unsigned long long* keys   = (unsigned long long*)(ws + WS_KEYS);
    int*                tpos   = (int*)(ws + WS_TPOS);
    float*              S      = (float*)(ws + WS_SPEC);

    k_init<<<2, 256, 0, stream>>>(keys, out);
    k_frames<<<BATCH * FPAD, 256, 0, stream>>>(recon, frames);

    dim3 gB(NSAMP / 128, 8, BATCH);          // 8 t-tiles/block, 8 frame-blocks, 4 batches
    k_conv_argmax<<<gB, 256, 0, stream>>>(frames, target, keys);

    k_positions<<<2, 256, 0, stream>>>(keys, tpos);

    dim3 gD(257, BATCH, 1);                  // ceil(2049 k-tiles / 8 waves)
    k_spectra<<<gD, 256, 0, stream>>>(frames, tpos, S);

    dim3 gE(NSAMP / 256, BATCH, 1);
    k_irfft_mse<<<gE, 256, 0, stream>>>(S, target, out);
}